// WaveletLayer_32693291057864
// MI455X (gfx1250) — compile-verified
//
#include <hip/hip_runtime.h>
#include <hip/hip_bf16.h>

typedef __attribute__((ext_vector_type(2))) float v2f;
typedef __attribute__((ext_vector_type(8))) float v8f;

#define INV_SQRT2 0.70710678118654752440f

// Problem geometry
constexpr int DD        = 44;   // feature dim
constexpr int KK        = 22;   // half dim (wavelet subband / GEMM K and N)
constexpr int TILE_ROWS = 16;   // rows (b,l) per wave  == WMMA M
constexpr int YL_STRIDE = 26;   // padded LDS stride for yl (even -> b64 aligned, 26%64 spreads banks)

__global__ __launch_bounds__(32)
void wavelet_ln_wmma_kernel(const float* __restrict__ x,     // (B*L, 44)
                            const float* __restrict__ Wlow,  // (22, 22)
                            const float* __restrict__ blow,  // (22,)
                            const float* __restrict__ lnw,   // (44,)
                            const float* __restrict__ lnb,   // (44,)
                            float* __restrict__ out,         // (B*L, 44)
                            int nrows)
{
    __shared__ float x_lds [TILE_ROWS * DD];        // raw input tile (stride 44)
    __shared__ float yl_lds[TILE_ROWS * YL_STRIDE]; // low-band, K padded 22->24 w/ zeros
    __shared__ float ylt_lds[TILE_ROWS * KK];       // GEMM result (stride 22)

    const int  lane = threadIdx.x;          // 0..31, wave32
    const long tile = blockIdx.x;
    const long row0 = tile * TILE_ROWS;

    // ---- 1) Stage 16 contiguous rows (704 fp32) global -> LDS, coalesced ----
    {
        const float* xg = x + row0 * DD;
        #pragma unroll
        for (int t = 0; t < 22; ++t) {
            int i = lane + 32 * t;          // 0..703
            x_lds[i] = xg[i];
        }
    }
    __syncthreads();

    // ---- 2) yl = (xe + xo) * invsqrt2 into padded LDS tile ----
    #pragma unroll
    for (int t = 0; t < 11; ++t) {          // 352 = 11 * 32 values
        int i = lane + 32 * t;
        int r = i / KK;
        int k = i - r * KK;
        float xe = x_lds[r * DD + 2 * k];
        float xo = x_lds[r * DD + 2 * k + 1];
        yl_lds[r * YL_STRIDE + k] = (xe + xo) * INV_SQRT2;
    }
    {   // zero the K padding (k = 22, 23) read by the last WMMA K-step
        int r = lane & 15;
        int k = KK + (lane >> 4);
        yl_lds[r * YL_STRIDE + k] = 0.0f;
    }
    __syncthreads();

    // ---- 3) B fragments (transposed W_low) + bias-initialized accumulators ----
    // B[k][n] = W_low[n][k].  16x4 fp32 B layout (mirror of A):
    //   lanes 0-15:  N=lane,    v0=K=k0,   v1=K=k0+1
    //   lanes 16-31: N=lane-16, v0=K=k0+2, v1=K=k0+3
    const int n0    = lane & 15;
    const int khalf = (lane >> 4) * 2;      // 0 or 2

    v2f bf0[6], bf1[6];
    #pragma unroll
    for (int s = 0; s < 6; ++s) {
        int k0 = 4 * s + khalf;
        int k1 = k0 + 1;
        v2f b0; v2f b1;
        b0.x = (k0 < KK) ? Wlow[n0 * KK + k0] : 0.0f;
        b0.y = (k1 < KK) ? Wlow[n0 * KK + k1] : 0.0f;
        int n1 = n0 + 16;                   // second N tile (cols 16..21 valid)
        b1.x = (n1 < KK && k0 < KK) ? Wlow[n1 * KK + k0] : 0.0f;
        b1.y = (n1 < KK && k1 < KK) ? Wlow[n1 * KK + k1] : 0.0f;
        bf0[s] = b0;
        bf1[s] = b1;
    }
    const float bias0 = blow[n0];
    const float bias1 = (n0 + 16 < KK) ? blow[n0 + 16] : 0.0f;

    v8f c0, c1;
    #pragma unroll
    for (int r = 0; r < 8; ++r) { c0[r] = bias0; c1[r] = bias1; }

    // ---- 4) GEMM: 6 K-steps x 2 N-tiles of v_wmma_f32_16x16x4_f32 ----
    // A fragment (16x4 fp32): lanes 0-15: M=lane, K=k0..k0+1 ; lanes 16-31: M=lane-16, K=k0+2..k0+3
    #pragma unroll
    for (int s = 0; s < 6; ++s) {
        int ka = 4 * s + khalf;             // even -> 8B-aligned ds_load_b64
        v2f a = *(const v2f*)(&yl_lds[n0 * YL_STRIDE + ka]);
        c0 = __builtin_amdgcn_wmma_f32_16x16x4_f32(false, a, false, bf0[s],
                                                   (short)0, c0, false, false);
        c1 = __builtin_amdgcn_wmma_f32_16x16x4_f32(false, a, false, bf1[s],
                                                   (short)0, c1, false, false);
    }

    // ---- 5) Scatter D tiles to LDS (C/D layout: VGPR r -> M=r (lanes<16) / M=r+8) ----
    const int mbase = (lane < 16) ? 0 : 8;
    #pragma unroll
    for (int r = 0; r < 8; ++r) {
        ylt_lds[(mbase + r) * KK + n0] = c0[r];
    }
    if (n0 < KK - 16) {                     // cols 16..21 of the second tile
        #pragma unroll
        for (int r = 0; r < 8; ++r) {
            ylt_lds[(mbase + r) * KK + n0 + 16] = c1[r];
        }
    }
    __syncthreads();

    // ---- 6) Inverse butterfly + residual + LayerNorm (2 lanes per row) ----
    const int row  = lane >> 1;             // 0..15
    const int half = lane & 1;
    const int j0   = half * 11;             // 11 subband cols per lane

    float yv[22];
    float sum = 0.0f;
    #pragma unroll
    for (int t = 0; t < 11; ++t) {
        int j = j0 + t;
        float ylt = ylt_lds[row * KK + j];
        v2f  xp  = *(const v2f*)(&x_lds[row * DD + 2 * j]);
        float yh = (xp.x - xp.y) * INV_SQRT2;
        float re = (ylt + yh) * INV_SQRT2;
        float ro = (ylt - yh) * INV_SQRT2;
        float y0 = re + xp.x;
        float y1 = ro + xp.y;
        yv[2 * t]     = y0;
        yv[2 * t + 1] = y1;
        sum += y0 + y1;
    }
    sum += __shfl_xor(sum, 1, 32);
    const float mean = sum * (1.0f / 44.0f);

    float s2 = 0.0f;
    #pragma unroll
    for (int t = 0; t < 22; ++t) {
        float d = yv[t] - mean;
        s2 += d * d;
    }
    s2 += __shfl_xor(s2, 1, 32);
    const float var  = s2 * (1.0f / 44.0f);
    const float rstd = 1.0f / sqrtf(var + 1e-12f);

    if (row0 + row < nrows) {
        float* og = out + (row0 + row) * DD;
        #pragma unroll
        for (int t = 0; t < 11; ++t) {
            int c = 2 * (j0 + t);
            v2f o;
            o.x = lnw[c]     * ((yv[2 * t]     - mean) * rstd) + lnb[c];
            o.y = lnw[c + 1] * ((yv[2 * t + 1] - mean) * rstd) + lnb[c + 1];
            *(v2f*)(&og[c]) = o;            // row*44 + 2j is 8B-aligned
        }
    }
}

extern "C" void kernel_launch(void* const* d_in, const int* in_sizes, int n_in,
                              void* d_out, int out_size, void* d_ws, size_t ws_size,
                              hipStream_t stream) {
    (void)n_in; (void)out_size; (void)d_ws; (void)ws_size;
    const float* x    = (const float*)d_in[0];   // (256,2048,44) fp32
    const float* Wlow = (const float*)d_in[1];   // (22,22)
    const float* blow = (const float*)d_in[2];   // (22,)
    const float* lnw  = (const float*)d_in[3];   // (44,)
    const float* lnb  = (const float*)d_in[4];   // (44,)
    float*       out  = (float*)d_out;

    const int total = in_sizes[0];               // B*L*44
    const int nrows = total / DD;                // 524288
    const int tiles = (nrows + TILE_ROWS - 1) / TILE_ROWS;  // 32768

    wavelet_ln_wmma_kernel<<<dim3(tiles), dim3(32), 0, stream>>>(
        x, Wlow, blow, lnw, lnb, out, nrows);
}